// Loss_84945863180678
// MI455X (gfx1250) — compile-verified
//
#include <hip/hip_runtime.h>

typedef float v2f __attribute__((ext_vector_type(2)));
typedef float v8f __attribute__((ext_vector_type(8)));

#define NN 8192
#define VV 8
#define OO 16
#define HH 128
#define INW (4 * VV + 3 * OO)   // 80 floats per scene
#define STEPSZ 2.0f
#define DIST_COST 1.0f
#define OBST_COST 0.1f
#define VEH_COST 0.1f

__device__ __forceinline__ float fcos(float x) {
    return __builtin_amdgcn_cosf(x * 0.15915494309189535f);  // v_cos_f32 takes revolutions
}
__device__ __forceinline__ float fsin(float x) {
    return __builtin_amdgcn_sinf(x * 0.15915494309189535f);
}

__global__ void zero_acc_kernel(float* gAcc, unsigned* gCnt) {
    gAcc[0] = 0.0f; gAcc[1] = 0.0f; gAcc[2] = 0.0f;
    gCnt[0] = 0u;   gCnt[1] = 0u;
}

__global__ __launch_bounds__(256) void traj_loss_kernel(
    const float* __restrict__ pred,     // (N, V, H, 1)
    const float* __restrict__ inputs,   // (N, 80)
    float* __restrict__ gAcc,           // [distSum, obstInvSum, vehInvSum]
    unsigned* __restrict__ gCnt)        // [obstCnt, vehCnt]
{
    __shared__ float sTraj[VV][HH][2];  // 8 KB trajectory cache
    __shared__ float sIn[INW];
    __shared__ float sAcc[3];
    __shared__ unsigned sCnt[2];

    const int tid = threadIdx.x;
    const int n = blockIdx.x;
    const int wid = tid >> 5;
    const int lane = tid & 31;

    if (tid < INW) sIn[tid] = inputs[(size_t)n * INW + tid];
    if (tid < 3) sAcc[tid] = 0.0f;
    if (tid < 2) sCnt[tid] = 0u;
    __syncthreads();

    float distS = 0.0f, oS = 0.0f, vS = 0.0f;
    unsigned ocnt = 0u, vcnt = 0u;

    // ---- 1) Trajectory cumsum: wave `wid` handles vehicle v=wid, 4 h per lane ----
    {
        const int v = wid;
        const float bx = sIn[4 * v + 0], by = sIn[4 * v + 1];
        const float tgx = sIn[4 * v + 2], tgy = sIn[4 * v + 3];

        const float4 ang = *(const float4*)(pred + (((size_t)n * VV + v) * HH + lane * 4));

        const float c0 = STEPSZ * fcos(ang.x), s0 = STEPSZ * fsin(ang.x);
        const float c1 = STEPSZ * fcos(ang.y), s1 = STEPSZ * fsin(ang.y);
        const float c2 = STEPSZ * fcos(ang.z), s2 = STEPSZ * fsin(ang.z);
        const float c3 = STEPSZ * fcos(ang.w), s3 = STEPSZ * fsin(ang.w);

        // lane-local inclusive prefix over the 4 elements
        const float px0 = c0, px1 = px0 + c1, px2 = px1 + c2, px3 = px2 + c3;
        const float py0 = s0, py1 = py0 + s1, py2 = py1 + s2, py3 = py2 + s3;

        // wave32 inclusive scan of lane totals
        float sx = px3, sy = py3;
        #pragma unroll
        for (int off = 1; off < 32; off <<= 1) {
            const float tx = __shfl_up(sx, off, 32);
            const float ty = __shfl_up(sy, off, 32);
            if (lane >= off) { sx += tx; sy += ty; }
        }
        const float ex = bx + (sx - px3);   // exclusive prefix + base
        const float ey = by + (sy - py3);

        const float X0 = ex + px0, Y0 = ey + py0;
        const float X1 = ex + px1, Y1 = ey + py1;
        const float X2 = ex + px2, Y2 = ey + py2;
        const float X3 = ex + px3, Y3 = ey + py3;

        const int h0 = lane * 4;
        sTraj[v][h0 + 0][0] = X0; sTraj[v][h0 + 0][1] = Y0;
        sTraj[v][h0 + 1][0] = X1; sTraj[v][h0 + 1][1] = Y1;
        sTraj[v][h0 + 2][0] = X2; sTraj[v][h0 + 2][1] = Y2;
        sTraj[v][h0 + 3][0] = X3; sTraj[v][h0 + 3][1] = Y3;

        // distance-to-target term (needs true sqrt)
        {
            float dx, dy;
            dx = tgx - X0; dy = tgy - Y0; distS += __builtin_amdgcn_sqrtf(dx * dx + dy * dy);
            dx = tgx - X1; dy = tgy - Y1; distS += __builtin_amdgcn_sqrtf(dx * dx + dy * dy);
            dx = tgx - X2; dy = tgy - Y2; distS += __builtin_amdgcn_sqrtf(dx * dx + dy * dy);
            dx = tgx - X3; dy = tgy - Y3; distS += __builtin_amdgcn_sqrtf(dx * dx + dy * dy);
        }
    }
    __syncthreads();

    // ---- 2) Obstacle term via V_WMMA_F32_16X16X4_F32 ----
    // d^2 = |p|^2 + |o|^2 - 2 p.o  as A(16x4) x B(4x16):
    //   A row m  = (px, py, |p|^2, 1)
    //   B col nn = (-2ox, -2oy, 1, |o|^2)
    {
        const int v = wid;
        const int col = lane & 15;
        const float obx = sIn[4 * VV + 3 * col + 0];
        const float oby = sIn[4 * VV + 3 * col + 1];
        const float obr = sIn[4 * VV + 3 * col + 2];
        const float R = obr + 1.0f;          // OBST_RADIUS
        const float R2 = R * R;              // mask: d^2 < R^2  (lane-uniform across D rows)

        v2f b;   // B 4x16: VGPR0 = rows K0(l<16)/K1(l>=16), VGPR1 = rows K2/K3
        b.x = (lane < 16) ? (-2.0f * obx) : (-2.0f * oby);
        b.y = (lane < 16) ? 1.0f : (obx * obx + oby * oby);

        #pragma unroll
        for (int t = 0; t < HH / 16; ++t) {
            const float px = sTraj[v][t * 16 + col][0];
            const float py = sTraj[v][t * 16 + col][1];
            v2f a;  // A 16x4: VGPR0 = K0(l<16)/K2(l>=16), VGPR1 = K1/K3
            a.x = (lane < 16) ? px : (px * px + py * py);
            a.y = (lane < 16) ? py : 1.0f;

            v8f c = {};
            const v8f d2 = __builtin_amdgcn_wmma_f32_16x16x4_f32(
                false, a, false, b, (short)0, c, false, false);

            #pragma unroll
            for (int j = 0; j < 8; ++j) {
                const float q = d2[j];
                const float inv = __builtin_amdgcn_rsqf(q);   // 1/d = rsqrt(d^2)
                const bool mk = q < R2;
                oS += mk ? inv : 0.0f;
                ocnt += mk ? 1u : 0u;
            }
        }
    }

    // ---- 3) Adjacent-vehicle term: 7*128 pairs, strided over the block ----
    for (int i = tid; i < (VV - 1) * HH; i += 256) {
        const int vv = 1 + (i / HH);
        const int h = i % HH;
        const float dx = sTraj[vv][h][0] - sTraj[vv - 1][h][0];
        const float dy = sTraj[vv][h][1] - sTraj[vv - 1][h][1];
        const float q = dx * dx + dy * dy;
        const bool mk = q < 1.0f;            // VEH_RADIUS^2
        const float inv = __builtin_amdgcn_rsqf(q);
        vS += mk ? inv : 0.0f;
        vcnt += mk ? 1u : 0u;
    }

    // ---- 4) Block reduce (LDS ds_add atomics), then one global atomic per block ----
    atomicAdd(&sAcc[0], distS);
    atomicAdd(&sAcc[1], oS);
    atomicAdd(&sAcc[2], vS);
    atomicAdd(&sCnt[0], ocnt);
    atomicAdd(&sCnt[1], vcnt);
    __syncthreads();
    if (tid == 0) {
        atomicAdd(&gAcc[0], sAcc[0]);
        atomicAdd(&gAcc[1], sAcc[1]);
        atomicAdd(&gAcc[2], sAcc[2]);
        atomicAdd(&gCnt[0], sCnt[0]);
        atomicAdd(&gCnt[1], sCnt[1]);
    }
}

__global__ void finalize_kernel(const float* gAcc, const unsigned* gCnt, float* out) {
    const float invPts = 1.0f / (float)((size_t)NN * VV * HH);
    float loss = gAcc[0] * invPts * DIST_COST;
    loss += (gCnt[0] > 0u) ? (gAcc[1] / (float)gCnt[0]) * OBST_COST : 0.0f;
    loss += (gCnt[1] > 0u) ? (gAcc[2] / (float)gCnt[1]) * VEH_COST : 0.0f;
    out[0] = loss;
}

extern "C" void kernel_launch(void* const* d_in, const int* in_sizes, int n_in,
                              void* d_out, int out_size, void* d_ws, size_t ws_size,
                              hipStream_t stream) {
    (void)in_sizes; (void)n_in; (void)out_size; (void)ws_size;
    const float* pred = (const float*)d_in[0];   // prediction (N,V,H,1) f32
    const float* inp  = (const float*)d_in[1];   // inputs (N,80) f32
    float* gAcc = (float*)d_ws;                                  // 3 floats
    unsigned* gCnt = (unsigned*)((char*)d_ws + 3 * sizeof(float)); // 2 u32
    float* out = (float*)d_out;

    zero_acc_kernel<<<1, 1, 0, stream>>>(gAcc, gCnt);
    traj_loss_kernel<<<NN, 256, 0, stream>>>(pred, inp, gAcc, gCnt);
    finalize_kernel<<<1, 1, 0, stream>>>(gAcc, gCnt, out);
}